// PatchAggregator_27599459844360
// MI455X (gfx1250) — compile-verified
//
#include <hip/hip_runtime.h>
#include <stdint.h>

// 7x7 joint-bilateral patch aggregation (B=4, C=3, Cf=8, H=W=256).
// Compute-bound fp32 stencil. CDNA5 paths: async global->LDS staging
// (global_load_async_to_lds_b32, ASYNCcnt) with zero-filled halo so the
// 49-tap compute loop needs zero bounds checks; v_rcp_f32 / v_exp_f32 for
// the chi^2 and weight math. WMMA intentionally absent: no shared-inner-dim
// matmul exists in this stencil (per-pixel-varying operands + per-element
// division + exp), so tensor ops would add no useful FLOPs.

#define IMG_H 256
#define IMG_W 256
#define CI 3
#define CF 8
#define NCH (CI + CF)            // 11
#define TILE 32
#define RAD 3
#define SM (TILE + 2 * RAD)      // 38
#define PLANE (SM * SM)          // 1444
#define SMEM_ELEMS (NCH * PLANE) // 15884 floats = 63,536 B

__global__ __launch_bounds__(256)
void patch_aggregate_kernel(const float* __restrict__ img,
                            const float* __restrict__ feat,
                            float* __restrict__ out)
{
    __shared__ float smem[SMEM_ELEMS];

    const int b   = blockIdx.z;
    const int gx0 = blockIdx.x * TILE;
    const int gy0 = blockIdx.y * TILE;
    const int tid = threadIdx.x;

    const float* imgB  = img  + (size_t)b * CI * IMG_H * IMG_W;
    const float* featB = feat + (size_t)b * CF * IMG_H * IMG_W;

    // ---- Stage halo tile (11 channels) into LDS via async copies;
    // ---- out-of-image halo cells are zero-filled (reference zero-pads). ----
    for (int idx = tid; idx < SMEM_ELEMS; idx += 256) {
        int ch = idx / PLANE;
        int r  = idx - ch * PLANE;
        int sy = r / SM;
        int sx = r - sy * SM;
        int gy = gy0 - RAD + sy;
        int gx = gx0 - RAD + sx;
        bool valid = ((unsigned)gy < (unsigned)IMG_H) &&
                     ((unsigned)gx < (unsigned)IMG_W);
        if (valid) {
            const float* src = (ch < CI)
                ? (imgB  + ((size_t)ch        * IMG_H + gy) * IMG_W + gx)
                : (featB + ((size_t)(ch - CI) * IMG_H + gy) * IMG_W + gx);
            uint32_t lds_addr = (uint32_t)(uintptr_t)(&smem[idx]);
            asm volatile("global_load_async_to_lds_b32 %0, %1, off"
                         :: "v"(lds_addr), "v"(src) : "memory");
        } else {
            smem[idx] = 0.0f;   // ds_store; DScnt settled by the barrier
        }
    }
    asm volatile("s_wait_asynccnt 0x0" ::: "memory");
    __syncthreads();

    // ---- Compute: 4 pixels per thread, 49 taps each, no bounds checks. ----
    // w = exp(-(dc/sc^2 + chi2/sf^2)) = exp2(dc*A + chi2*B)
    constexpr float A2 = (float)(-(1.0 / (0.1 * 0.1 + 1e-8)) * 1.4426950408889634);
    constexpr float B2 = (float)(-(1.0 / (0.2 * 0.2 + 1e-8)) * 1.4426950408889634);

    const int lx  = tid & 31;
    const int ly0 = tid >> 5;  // 0..7

#pragma unroll 1
    for (int i = 0; i < 4; ++i) {
        const int ly   = ly0 + 8 * i;
        const int cIdx = (ly + RAD) * SM + (lx + RAD);

        const float c0 = smem[0 * PLANE + cIdx];
        const float c1 = smem[1 * PLANE + cIdx];
        const float c2 = smem[2 * PLANE + cIdx];
        float fc[CF], fce[CF];
#pragma unroll
        for (int k = 0; k < CF; ++k) {
            fc[k]  = smem[(CI + k) * PLANE + cIdx];
            fce[k] = fc[k] + 1e-6f;
        }

        float num0 = 0.f, num1 = 0.f, num2 = 0.f, den = 0.f;

#pragma unroll 1
        for (int dy = -RAD; dy <= RAD; ++dy) {
            const int rowOff = cIdx + dy * SM;
#pragma unroll
            for (int dx = -RAD; dx <= RAD; ++dx) {
                const int o = rowOff + dx;

                const float p0 = smem[0 * PLANE + o];
                const float p1 = smem[1 * PLANE + o];
                const float p2 = smem[2 * PLANE + o];

                float d  = p0 - c0;
                float dc = d * d;
                d = p1 - c1; dc = fmaf(d, d, dc);
                d = p2 - c2; dc = fmaf(d, d, dc);

                float chi2 = 0.f;
#pragma unroll
                for (int k = 0; k < CF; ++k) {
                    const float pf = smem[(CI + k) * PLANE + o];
                    const float df = pf - fc[k];
                    const float s  = pf + fce[k];          // + 1e-6 folded in
                    chi2 = fmaf(df * df, __builtin_amdgcn_rcpf(s), chi2);
                }

                const float w = __builtin_amdgcn_exp2f(fmaf(chi2, B2, dc * A2));
                num0 = fmaf(w, p0, num0);
                num1 = fmaf(w, p1, num1);
                num2 = fmaf(w, p2, num2);
                den += w;
            }
        }

        const float rden = __builtin_amdgcn_rcpf(den + 1e-6f);
        const size_t px = ((size_t)gy0 + ly) * IMG_W + (gx0 + lx);
        float* outB = out + (size_t)b * CI * IMG_H * IMG_W;
        outB[0 * (size_t)IMG_H * IMG_W + px] = num0 * rden;
        outB[1 * (size_t)IMG_H * IMG_W + px] = num1 * rden;
        outB[2 * (size_t)IMG_H * IMG_W + px] = num2 * rden;
    }
}

extern "C" void kernel_launch(void* const* d_in, const int* in_sizes, int n_in,
                              void* d_out, int out_size, void* d_ws, size_t ws_size,
                              hipStream_t stream) {
    const float* img  = (const float*)d_in[0];  // [B,3,256,256]
    const float* feat = (const float*)d_in[1];  // [B,8,256,256]
    float* out = (float*)d_out;                 // [B,3,256,256]

    const int B = in_sizes[0] / (CI * IMG_H * IMG_W);  // = 4
    dim3 grid(IMG_W / TILE, IMG_H / TILE, B);
    dim3 block(256);
    patch_aggregate_kernel<<<grid, block, 0, stream>>>(img, feat, out);
}